// GCN_22333829939712
// MI455X (gfx1250) — compile-verified
//
#include <hip/hip_runtime.h>
#include <hip/hip_bf16.h>

// ---------------------------------------------------------------------------
// 3-layer GCN (PyG GCNConv semantics) for MI455X / gfx1250, wave32.
//   norm_e = deg^-1/2[src] * deg^-1/2[dst], deg includes self-loop (+1)
//   layer: h = x@W ; agg[d] = b + dinv[d]^2*h[d] + sum_e norm_e*h[src_e]
//   relu after layers 1,2 ; log_softmax after layer 3
// GEMMs: V_WMMA_F32_16X16X4_F32 (full f32 precision, matches reference).
// Aggregation: coalesced per-lane edge loads + v_readlane broadcast, then
// float4 gathers and hardware f32 atomics resolving in the 192MB L2.
// ---------------------------------------------------------------------------

typedef __attribute__((ext_vector_type(2))) float v2f;
typedef __attribute__((ext_vector_type(8))) float v8f;

#define NNODES 50000
#define F_IN   256
#define F_HID  128
#define F_OUT  40

__device__ __forceinline__ float readlane_f32(float v, int j) {
  return __int_as_float(__builtin_amdgcn_readlane(__float_as_int(v), j));
}

// ---------------- degree / norm ----------------

__global__ void k_fill(float* __restrict__ p, float v, int n) {
  int i = blockIdx.x * blockDim.x + threadIdx.x;
  for (; i < n; i += gridDim.x * blockDim.x) p[i] = v;
}

__global__ void k_deg_edges(const int* __restrict__ ei, float* __restrict__ deg, int nE) {
  int e = blockIdx.x * blockDim.x + threadIdx.x;
  if (e >= nE) return;
  int d = ei[nE + e];                      // dst row of edge_index
  unsafeAtomicAdd(&deg[d], 1.0f);
}

__global__ void k_rsqrt_inplace(float* __restrict__ deg, int n) {
  int i = blockIdx.x * blockDim.x + threadIdx.x;
  for (; i < n; i += gridDim.x * blockDim.x) deg[i] = __frsqrt_rn(deg[i]);
}

// ---------------- WMMA f32 GEMM: C[M x Nreal] = A[M x K] * B[K x Nreal] ----
// One wave computes one 16x16 output tile. M must be a multiple of 16.
// N tiles padded to 16; out-of-range columns clamped+masked so EXEC stays
// all-1s at the WMMA itself.

__global__ void k_gemm_wmma_f32(const float* __restrict__ A,
                                const float* __restrict__ B,
                                float* __restrict__ C,
                                int K, int Nreal, int nTilesN, int totalTiles) {
  int wid  = (blockIdx.x * blockDim.x + threadIdx.x) >> 5;
  if (wid >= totalTiles) return;          // wave-uniform exit
  int lane = threadIdx.x & 31;

  int tm = wid / nTilesN;
  int tn = wid - tm * nTilesN;

  int m  = lane & 15;                     // A: lane -> row
  int kh = (lane >> 4) << 1;              // half-wave -> K pair {0,1} or {2,3}
  int n  = (tn << 4) + (lane & 15);       // B/C: lane -> col
  int ncl = (n < Nreal) ? n : (Nreal - 1);
  float bmask = (n < Nreal) ? 1.0f : 0.0f;

  const float* Ap = A + (size_t)(tm * 16 + m) * K + kh;   // 8B-aligned (kh even)
  const float* Bp = B + ncl;

  v8f acc = {};
  for (int kk = 0; kk < K; kk += 4) {
    float2 av = *(const float2*)(Ap + kk);
    v2f a; a.x = av.x; a.y = av.y;
    v2f b;
    b.x = Bp[(size_t)(kk + kh)     * Nreal] * bmask;
    b.y = Bp[(size_t)(kk + kh + 1) * Nreal] * bmask;
    acc = __builtin_amdgcn_wmma_f32_16x16x4_f32(
        /*neg_a=*/false, a, /*neg_b=*/false, b,
        /*c_mod=*/(short)0, acc, /*reuse_a=*/false, /*reuse_b=*/false);
  }

  if (n < Nreal) {
    int rbase = (lane >> 4) << 3;         // rows 0..7 or 8..15
    float* Cp = C + (size_t)(tm * 16 + rbase) * Nreal + n;
#pragma unroll
    for (int r = 0; r < 8; ++r)
      Cp[(size_t)r * Nreal] = acc[r];
  }
}

// ---------------- aggregation ----------------

// agg[i,:] = b + dinv[i]^2 * h[i,:]  (bias + self-loop), float4-vectorized.
// total4 = M * F / 4 ; F4 = F / 4.
__global__ void k_agg_init4(const float4* __restrict__ h, const float* __restrict__ dinv,
                            const float4* __restrict__ bias, float4* __restrict__ agg,
                            int F4, int total4) {
  int i = blockIdx.x * blockDim.x + threadIdx.x;
  for (; i < total4; i += gridDim.x * blockDim.x) {
    int row = i / F4, q = i - row * F4;
    float di = dinv[row];
    float w = di * di;
    float4 hv = h[i], bv = bias[q];
    float4 o;
    o.x = bv.x + w * hv.x;  o.y = bv.y + w * hv.y;
    o.z = bv.z + w * hv.z;  o.w = bv.w + w * hv.w;
    agg[i] = o;
  }
}

// One wave handles 32 edges: lane e-loads are fully coalesced, then each
// edge's (src,dst,norm) is broadcast to SGPRs via v_readlane; all 32 lanes
// cooperatively gather h[src] (float4 each = 128 feats) and atomically
// accumulate into agg[dst]. Tail edges get norm=0 (exact no-op adds).
__global__ void k_agg_edges_f128(const int* __restrict__ ei, const float* __restrict__ h,
                                 const float* __restrict__ dinv, float* __restrict__ agg,
                                 int nE) {
  int wid  = (blockIdx.x * blockDim.x + threadIdx.x) >> 5;
  int lane = threadIdx.x & 31;
  int base = wid << 5;
  if (base >= nE) return;
  int e = base + lane;
  bool valid = e < nE;
  int   sv = valid ? ei[e]      : 0;
  int   dv = valid ? ei[nE + e] : 0;
  float nv = valid ? dinv[sv] * dinv[dv] : 0.0f;
#pragma unroll 4
  for (int j = 0; j < 32; ++j) {
    int   s   = __builtin_amdgcn_readlane(sv, j);
    int   d   = __builtin_amdgcn_readlane(dv, j);
    float nrm = readlane_f32(nv, j);
    float4 v = *(const float4*)(h + (size_t)s * 128 + lane * 4);
    float* op = agg + (size_t)d * 128 + lane * 4;
    unsafeAtomicAdd(op + 0, v.x * nrm);
    unsafeAtomicAdd(op + 1, v.y * nrm);
    unsafeAtomicAdd(op + 2, v.z * nrm);
    unsafeAtomicAdd(op + 3, v.w * nrm);
  }
}

// Same scheme for F = 40: lane covers f=lane and f=lane+32 (lane<8).
__global__ void k_agg_edges_f40(const int* __restrict__ ei, const float* __restrict__ h,
                                const float* __restrict__ dinv, float* __restrict__ agg,
                                int nE) {
  int wid  = (blockIdx.x * blockDim.x + threadIdx.x) >> 5;
  int lane = threadIdx.x & 31;
  int base = wid << 5;
  if (base >= nE) return;
  int e = base + lane;
  bool valid = e < nE;
  int   sv = valid ? ei[e]      : 0;
  int   dv = valid ? ei[nE + e] : 0;
  float nv = valid ? dinv[sv] * dinv[dv] : 0.0f;
#pragma unroll 4
  for (int j = 0; j < 32; ++j) {
    int   s   = __builtin_amdgcn_readlane(sv, j);
    int   d   = __builtin_amdgcn_readlane(dv, j);
    float nrm = readlane_f32(nv, j);
    const float* hp = h + (size_t)s * 40;
    float* op = agg + (size_t)d * 40;
    unsafeAtomicAdd(op + lane, hp[lane] * nrm);
    if (lane < 8) unsafeAtomicAdd(op + lane + 32, hp[lane + 32] * nrm);
  }
}

__global__ void k_relu4(float4* __restrict__ p, int n4) {
  int i = blockIdx.x * blockDim.x + threadIdx.x;
  for (; i < n4; i += gridDim.x * blockDim.x) {
    float4 v = p[i];
    v.x = fmaxf(v.x, 0.0f); v.y = fmaxf(v.y, 0.0f);
    v.z = fmaxf(v.z, 0.0f); v.w = fmaxf(v.w, 0.0f);
    p[i] = v;
  }
}

// ---------------- log_softmax over 40 cols, one wave per row --------------

__global__ void k_logsoftmax40(float* __restrict__ out, int M) {
  int wid  = (blockIdx.x * blockDim.x + threadIdx.x) >> 5;
  int lane = threadIdx.x & 31;
  if (wid >= M) return;
  float* z = out + (size_t)wid * 40;
  float a0 = z[lane];
  float a1 = (lane < 8) ? z[lane + 32] : -3.402823466e38f;
  float mx = fmaxf(a0, a1);
#pragma unroll
  for (int off = 16; off >= 1; off >>= 1)
    mx = fmaxf(mx, __shfl_xor(mx, off, 32));
  float s = expf(a0 - mx) + ((lane < 8) ? expf(a1 - mx) : 0.0f);
#pragma unroll
  for (int off = 16; off >= 1; off >>= 1)
    s += __shfl_xor(s, off, 32);
  float lse = mx + logf(s);
  z[lane] = a0 - lse;
  if (lane < 8) z[lane + 32] = a1 - lse;
}

// ---------------- driver ----------------

extern "C" void kernel_launch(void* const* d_in, const int* in_sizes, int n_in,
                              void* d_out, int out_size, void* d_ws, size_t ws_size,
                              hipStream_t stream) {
  const float* x  = (const float*)d_in[0];
  const int*   ei = (const int*)d_in[1];          // [2, nE], int32
  const float* W1 = (const float*)d_in[2];
  const float* b1 = (const float*)d_in[3];
  const float* W2 = (const float*)d_in[4];
  const float* b2 = (const float*)d_in[5];
  const float* W3 = (const float*)d_in[6];
  const float* b3 = (const float*)d_in[7];
  float* out = (float*)d_out;

  const int nE = in_sizes[1] / 2;                 // 1,600,000
  const int M  = NNODES;                          // 50,000 (multiple of 16)

  // workspace layout (512B-aligned regions)
  char* ws = (char*)d_ws;
  float* dinv = (float*)ws;                       // 50000 f32 (deg -> dinv)
  float* h    = (float*)(ws + 204800);            // 50000x128 f32 (also 50000x40)
  float* agg  = (float*)(ws + 204800 + 25600000); // 50000x128 f32

  const int B256 = 256;
  const int edgeWaves  = (nE + 31) / 32;
  const int edgeBlocks = (edgeWaves * 32 + B256 - 1) / B256;

  // --- degrees / symmetric norm ---
  k_fill<<<(M + B256 - 1) / B256, B256, 0, stream>>>(dinv, 1.0f, M);  // self-loop
  k_deg_edges<<<(nE + B256 - 1) / B256, B256, 0, stream>>>(ei, dinv, nE);
  k_rsqrt_inplace<<<(M + B256 - 1) / B256, B256, 0, stream>>>(dinv, M);

  const int mTiles = M / 16;                      // 3125

  // --- layer 1: h = x @ W1 (K=256, N=128) ---
  {
    int nT = F_HID / 16, total = mTiles * nT;
    k_gemm_wmma_f32<<<(total * 32 + B256 - 1) / B256, B256, 0, stream>>>(
        x, W1, h, F_IN, F_HID, nT, total);
  }
  k_agg_init4<<<(M * 32 + B256 - 1) / B256, B256, 0, stream>>>(
      (const float4*)h, dinv, (const float4*)b1, (float4*)agg, F_HID / 4, M * F_HID / 4);
  k_agg_edges_f128<<<edgeBlocks, B256, 0, stream>>>(ei, h, dinv, agg, nE);
  k_relu4<<<(M * 32 + B256 - 1) / B256, B256, 0, stream>>>((float4*)agg, M * F_HID / 4);

  // --- layer 2: h = agg @ W2 (K=128, N=128) ---
  {
    int nT = F_HID / 16, total = mTiles * nT;
    k_gemm_wmma_f32<<<(total * 32 + B256 - 1) / B256, B256, 0, stream>>>(
        agg, W2, h, F_HID, F_HID, nT, total);
  }
  k_agg_init4<<<(M * 32 + B256 - 1) / B256, B256, 0, stream>>>(
      (const float4*)h, dinv, (const float4*)b2, (float4*)agg, F_HID / 4, M * F_HID / 4);
  k_agg_edges_f128<<<edgeBlocks, B256, 0, stream>>>(ei, h, dinv, agg, nE);
  k_relu4<<<(M * 32 + B256 - 1) / B256, B256, 0, stream>>>((float4*)agg, M * F_HID / 4);

  // --- layer 3: h3 = agg @ W3 (K=128, N=40 padded to 48), aggregate into d_out ---
  {
    int nT = 3, total = mTiles * nT;              // 48 padded cols
    k_gemm_wmma_f32<<<(total * 32 + B256 - 1) / B256, B256, 0, stream>>>(
        agg, W3, h, F_HID, F_OUT, nT, total);     // h reused as 50000x40
  }
  k_agg_init4<<<(M * 10 + B256 - 1) / B256, B256, 0, stream>>>(
      (const float4*)h, dinv, (const float4*)b3, (float4*)out, F_OUT / 4, M * F_OUT / 4);
  k_agg_edges_f40<<<edgeBlocks, B256, 0, stream>>>(ei, h, dinv, out, nE);

  // --- log_softmax in place on d_out ---
  k_logsoftmax40<<<(M * 32 + B256 - 1) / B256, B256, 0, stream>>>(out, M);
}